// EuclideanLoss2_25658134626536
// MI455X (gfx1250) — compile-verified
//
#include <hip/hip_runtime.h>
#include <math.h>

typedef float v2f __attribute__((ext_vector_type(2)));
typedef float v8f __attribute__((ext_vector_type(8)));

// Problem constants from the reference: x[B,N,D], y[B,D,N], D=3, output scalar.
#define BB 8
#define NN 4096
#define DD 3
#define TILES_PER_B (NN / 16)        // 256
#define NUM_TILES   (BB * TILES_PER_B) // 2048

// Pass 1: one wave per 16-row diagonal tile of one batch.
// Uses V_WMMA_F32_16X16X4_F32 to compute the 16x16 product of
// A = x[b, i0:i0+16, 0:4] (K=3 padded to 4 with zeros) and
// B = y[b, 0:4, i0:i0+16]; only the diagonal is consumed.
__global__ void __launch_bounds__(256)
diag_dist_pass1(const float* __restrict__ x,
                const float* __restrict__ y,
                float* __restrict__ partial) {
  const int gtid = blockIdx.x * blockDim.x + threadIdx.x;
  const int wave = gtid >> 5;          // global wave id == tile id
  const int lane = threadIdx.x & 31;
  const int b    = wave >> 8;          // 256 tiles per batch
  const int i0   = (wave & 255) << 4;  // tile row base

  const float* xb = x + (size_t)b * NN * DD;   // rows of length 3
  const float* yb = y + (size_t)b * DD * NN;   // 3 rows of length N

  // A fragment (16x4 f32): lanes 0-15 -> K=0,1 ; lanes 16-31 -> K=2,3(pad 0)
  // B fragment (4x16 f32): lanes 0-15 -> rows K=0,1 ; lanes 16-31 -> K=2,3(pad 0)
  v2f afrag, bfrag;
  if (lane < 16) {
    const int r = i0 + lane;
    afrag.x = xb[r * 3 + 0];
    afrag.y = xb[r * 3 + 1];
    bfrag.x = yb[0 * NN + r];
    bfrag.y = yb[1 * NN + r];
  } else {
    const int r = i0 + lane - 16;
    afrag.x = xb[r * 3 + 2];
    afrag.y = 0.0f;
    bfrag.x = yb[2 * NN + r];
    bfrag.y = 0.0f;
  }

  v8f c = {};
  // 8 args: (neg_a, A, neg_b, B, c_mod, C, reuse_a, reuse_b)
  c = __builtin_amdgcn_wmma_f32_16x16x4_f32(
      false, afrag, false, bfrag, (short)0, c, false, false);

  // Diagonal extraction per C/D layout:
  //   m in 0..7  -> lane m      , VGPR m
  //   m in 8..15 -> lane m + 16 , VGPR m - 8
  const bool own  = (lane < 8) || (lane >= 24);
  const int  m    = (lane < 8) ? lane : (lane - 16);
  const int  cidx = (lane < 8) ? lane : (lane - 24);
  float cv = c[0];
#pragma unroll
  for (int k = 1; k < 8; ++k) cv = (cidx == k) ? c[k] : cv;

  float val = 0.0f;
  if (own) {
    const int i = i0 + m;
    const float x0 = xb[i * 3 + 0], x1 = xb[i * 3 + 1], x2 = xb[i * 3 + 2];
    const float y0 = yb[i], y1 = yb[NN + i], y2 = yb[2 * NN + i];
    const float a2 = x0 * x0 + x1 * x1 + x2 * x2;
    const float b2 = y0 * y0 + y1 * y1 + y2 * y2;
    const float d2 = a2 + b2 - 2.0f * cv;          // matches reference formula
    const float dv = sqrtf(fmaxf(d2, 0.0f));
    val = ((i == 1) || (i == 2)) ? 1.5f * dv : dv; // fold diag[1:3] *= 1.5
  }

  // Full wave32 reduction (deterministic butterfly).
#pragma unroll
  for (int off = 16; off > 0; off >>= 1)
    val += __shfl_xor(val, off, 32);

  if (lane == 0) partial[wave] = val;
}

// Pass 2: single block, fixed-order tree reduction -> bit-deterministic.
__global__ void __launch_bounds__(256)
diag_dist_pass2(const float* __restrict__ partial, float* __restrict__ out) {
  __shared__ float smem[256];
  float v = 0.0f;
  for (int idx = threadIdx.x; idx < NUM_TILES; idx += 256) v += partial[idx];
  smem[threadIdx.x] = v;
  __syncthreads();
#pragma unroll
  for (int off = 128; off > 0; off >>= 1) {
    if ((int)threadIdx.x < off) smem[threadIdx.x] += smem[threadIdx.x + off];
    __syncthreads();
  }
  if (threadIdx.x == 0)
    out[0] = smem[0] * (1.0f / (float)(BB * NN)); // mean over B then over N
}

extern "C" void kernel_launch(void* const* d_in, const int* in_sizes, int n_in,
                              void* d_out, int out_size, void* d_ws, size_t ws_size,
                              hipStream_t stream) {
  const float* x = (const float*)d_in[0];  // [B,N,D]
  const float* y = (const float*)d_in[1];  // [B,D,N]
  (void)d_in; (void)in_sizes; (void)n_in;  // d_in[2] = alt is dead code in the reference
  float* out     = (float*)d_out;
  float* partial = (float*)d_ws;           // NUM_TILES floats = 8 KB

  // 2048 waves -> 256 blocks of 8 waves (256 threads, wave32).
  diag_dist_pass1<<<NUM_TILES / 8, 256, 0, stream>>>(x, y, partial);
  diag_dist_pass2<<<1, 256, 0, stream>>>(partial, out);
  (void)out_size; (void)ws_size;
}